// DiscreteCDELayer_41601053229407
// MI455X (gfx1250) — compile-verified
//
#include <hip/hip_runtime.h>

// ---------------------------------------------------------------------------
// Discrete CDE scan, persistent-workgroup fp32 WMMA kernel for gfx1250.
//   B=256, T=1024, H=128, I=32, W=128, H*I=4096
// 16 workgroups (one per 16-batch-row tile), 8 wave32s each, loop over T.
// GEMMs via V_WMMA_F32_16X16X4_F32; y/h1/h2/dx tiles live in LDS (26 KB).
// W1/W2/W3/b* are read from global (hot in L2: W3 = 2MB << 192MB L2).
// ---------------------------------------------------------------------------

typedef float v2f __attribute__((ext_vector_type(2)));
typedef float v8f __attribute__((ext_vector_type(8)));

#define BB   256
#define TT   1024
#define HH   128
#define II   32
#define MW   128          // MLP width
#define HI   4096         // H*I
#define BM   16           // batch rows per workgroup (one WMMA M tile)
#define NWAVES 8
#define NTHR (NWAVES * 32)

__device__ __forceinline__ v8f wmma4(v2f a, v2f b, v8f c) {
  // D = A(16x4 f32) * B(4x16 f32) + C(16x16 f32)
  return __builtin_amdgcn_wmma_f32_16x16x4_f32(
      /*neg_a=*/false, a, /*neg_b=*/false, b,
      /*c_mod=*/(short)0, c, /*reuse_a=*/false, /*reuse_b=*/false);
}

__device__ __forceinline__ float softplus_f(float x) {
  // stable: max(x,0) + log1p(exp(-|x|))
  return fmaxf(x, 0.0f) + log1pf(expf(-fabsf(x)));
}

__global__ __launch_bounds__(NTHR, 1)
void cde_scan_kernel(const float* __restrict__ y0,
                     const float* __restrict__ xs1,
                     const float* __restrict__ W1, const float* __restrict__ b1,
                     const float* __restrict__ W2, const float* __restrict__ b2,
                     const float* __restrict__ W3, const float* __restrict__ b3,
                     float* __restrict__ out) {
  __shared__ float sY [BM][HH];   // running state tile      (8 KB)
  __shared__ float sH1[BM][MW];   // softplus layer 1        (8 KB)
  __shared__ float sH2[BM][MW];   // softplus layer 2        (8 KB)
  __shared__ float sDX[BM][II];   // dx tile                 (2 KB)

  const int tid  = threadIdx.x;
  const int wave = tid >> 5;
  const int lane = tid & 31;
  const int l16  = lane & 15;
  const int hi16 = lane >> 4;        // 0 or 1
  const int m0   = blockIdx.x * BM;  // first batch row of this tile

  // ---- init y tile from y0 ----
  for (int e = tid; e < BM * HH; e += NTHR)
    sY[e >> 7][e & 127] = y0[(size_t)(m0 + (e >> 7)) * HH + (e & 127)];

  // per-wave constant biases for layers 1/2 (column = 16*wave + l16)
  const int   ncol = wave * 16 + l16;
  const float b1v  = b1[ncol];
  const float b2v  = b2[ncol];

  // dx staging: each thread owns 2 consecutive elements of the 16x32 tile
  const int de = tid * 2;            // 0..511
  const int dm = de >> 5;            // row in tile
  const int di = de & 31;            // input index (even)
  float px = 0.0f, py = 0.0f;        // x_{-1} = 0

  __syncthreads();

  for (int t = 0; t < TT; ++t) {
    // ---- stage dx = x_t - x_{t-1} into LDS ----
    {
      const float* xp = xs1 + ((size_t)(m0 + dm) * TT + t) * II + di;
      v2f cur = *(const v2f*)xp;
      sDX[dm][di]     = cur.x - px;
      sDX[dm][di + 1] = cur.y - py;
      px = cur.x; py = cur.y;
    }

    // ---- phase 1: h1 = softplus(y @ W1^T + b1), wave w -> cols [16w,16w+16) ----
    {
      v8f c = {0.f,0.f,0.f,0.f,0.f,0.f,0.f,0.f};
      const int nb = wave * 16;
#pragma unroll
      for (int kb = 0; kb < HH; kb += 4) {
        v2f a = *(const v2f*)&sY[l16][kb + 2 * hi16];
        v2f b = *(const v2f*)&W1[(size_t)(nb + l16) * HH + kb + 2 * hi16];
        c = wmma4(a, b, c);
      }
#pragma unroll
      for (int j = 0; j < 8; ++j)
        sH1[j + 8 * hi16][nb + l16] = softplus_f(c[j] + b1v);
    }
    __syncthreads();

    // ---- phase 2: h2 = softplus(h1 @ W2^T + b2) ----
    {
      v8f c = {0.f,0.f,0.f,0.f,0.f,0.f,0.f,0.f};
      const int nb = wave * 16;
#pragma unroll
      for (int kb = 0; kb < MW; kb += 4) {
        v2f a = *(const v2f*)&sH1[l16][kb + 2 * hi16];
        v2f b = *(const v2f*)&W2[(size_t)(nb + l16) * MW + kb + 2 * hi16];
        c = wmma4(a, b, c);
      }
#pragma unroll
      for (int j = 0; j < 8; ++j)
        sH2[j + 8 * hi16][nb + l16] = softplus_f(c[j] + b2v);
    }
    __syncthreads();

    // ---- phase 3: g = tanh(h2 @ W3^T + b3); y += einsum('hi,i->h', g, dx) ----
    {
      // hoist all 32 A k-fragments of h2 (reused by all 32 N-tiles)
      v2f af[32];
#pragma unroll
      for (int kk = 0; kk < 32; ++kk)
        af[kk] = *(const v2f*)&sH2[l16][4 * kk + 2 * hi16];

      // register-cache dx: lane needs dx[m][l16] and dx[m][16+l16], m = j+8*hi16
      float dxA[8], dxB[8];
#pragma unroll
      for (int j = 0; j < 8; ++j) {
        dxA[j] = sDX[j + 8 * hi16][l16];
        dxB[j] = sDX[j + 8 * hi16][16 + l16];
      }

      for (int hh = 0; hh < 16; ++hh) {          // h = 16*wave + hh
        const int h = wave * 16 + hh;
        float acc[8] = {0.f,0.f,0.f,0.f,0.f,0.f,0.f,0.f};
#pragma unroll
        for (int half = 0; half < 2; ++half) {   // i in [0,16) / [16,32)
          const int nb3 = h * II + 16 * half;    // column base in 4096
          v8f c = {0.f,0.f,0.f,0.f,0.f,0.f,0.f,0.f};
#pragma unroll
          for (int kk = 0; kk < 32; ++kk) {
            v2f b = *(const v2f*)&W3[(size_t)(nb3 + l16) * MW + 4 * kk + 2 * hi16];
            c = wmma4(af[kk], b, c);
          }
          const float b3v = b3[nb3 + l16];
#pragma unroll
          for (int j = 0; j < 8; ++j) {
            float g = tanhf(c[j] + b3v);
            acc[j] += g * (half ? dxB[j] : dxA[j]);
          }
        }
        // reduce over the 16 N-lanes (i dimension); masks<16 stay in-half
#pragma unroll
        for (int j = 0; j < 8; ++j) {
          float v = acc[j];
          v += __shfl_xor(v, 1, 32);
          v += __shfl_xor(v, 2, 32);
          v += __shfl_xor(v, 4, 32);
          v += __shfl_xor(v, 8, 32);
          acc[j] = v;
        }
        if (l16 == 0) {
#pragma unroll
          for (int j = 0; j < 8; ++j)
            sY[j + 8 * hi16][h] += acc[j];
        }
      }
    }
    __syncthreads();

    // ---- emit ys[t] = y ----
    {
      float* ysp = out + (size_t)BB * HH;  // ys region after yT
#pragma unroll
      for (int r = 0; r < 8; ++r) {
        const int e = r * NTHR + tid;      // 0..2047
        const int m = e >> 7, hc = e & 127;
        ysp[((size_t)t * BB + (m0 + m)) * HH + hc] = sY[m][hc];
      }
    }
  }

  // ---- emit yT ----
#pragma unroll
  for (int r = 0; r < 8; ++r) {
    const int e = r * NTHR + tid;
    const int m = e >> 7, hc = e & 127;
    out[(size_t)(m0 + m) * HH + hc] = sY[m][hc];
  }
}

extern "C" void kernel_launch(void* const* d_in, const int* in_sizes, int n_in,
                              void* d_out, int out_size, void* d_ws, size_t ws_size,
                              hipStream_t stream) {
  (void)in_sizes; (void)n_in; (void)d_ws; (void)ws_size; (void)out_size;
  const float* y0  = (const float*)d_in[0];
  const float* xs1 = (const float*)d_in[1];
  const float* W1  = (const float*)d_in[2];
  const float* b1  = (const float*)d_in[3];
  const float* W2  = (const float*)d_in[4];
  const float* b2  = (const float*)d_in[5];
  const float* W3  = (const float*)d_in[6];
  const float* b3  = (const float*)d_in[7];

  dim3 grid(BB / BM);   // 16 persistent workgroups
  dim3 block(NTHR);     // 8 wave32s
  cde_scan_kernel<<<grid, block, 0, stream>>>(y0, xs1, W1, b1, W2, b2, W3, b3,
                                              (float*)d_out);
}